// GraphTransformer_61718680043593
// MI455X (gfx1250) — compile-verified
//
#include <hip/hip_runtime.h>

#define NNODES 32768
#define NEDGES 524288
#define HID 128
#define KBLK 64    // K-rows staged per LDS block
#define KP 68      // LDS k-pitch: KP%64==4 -> b64 frag loads are bank-conflict-free

typedef __attribute__((ext_vector_type(2))) float v2f;
typedef __attribute__((ext_vector_type(8))) float v8f;

// ---------------------------------------------------------------------------
// V_WMMA_F32_16X16X4_F32. Layout (wave32): hi = lane>>4, lm = lane&15.
//   A frag: a[r] = A[lm][2*hi + r]      B frag: b[r] = B[2*hi + r][lm]
//   C frag: c[r] = C[8*hi + r][lm]
// ---------------------------------------------------------------------------
__device__ __forceinline__ v8f wmma4(v2f a, v2f b, v8f c) {
  return __builtin_amdgcn_wmma_f32_16x16x4_f32(false, a, false, b, (short)0, c,
                                               false, false);
}

// Stage B[kb..kb+kcnt) x 128 cols (base nb) into LDS *transposed*:
// lds[n*KP + k]. Fragment (k, k+1) for one column is then one ds_load_b64.
__device__ __forceinline__ void stageBT(const float* __restrict__ B, int ldb,
                                        int nb, int kb, int kcnt, float* lds) {
  __syncthreads();  // previous block fully consumed
  for (int idx = threadIdx.x; idx < kcnt * 128; idx += 256) {
    const int k = idx >> 7, n = idx & 127;
    lds[n * KP + k] = B[(kb + k) * ldb + nb + n];
  }
  __syncthreads();
}

// Full-K GEMM of one 16x128 strip: c[t] += A_strip * B, B staged via LDS.
__device__ __forceinline__ void gemm_full(const float* __restrict__ Arow,
                                          const float* __restrict__ B, int ldb,
                                          int nb, int K, int hi, int lm,
                                          float* lds, v8f c[8]) {
  for (int kb = 0; kb < K; kb += KBLK) {
    const int kcnt = (K - kb < KBLK) ? (K - kb) : KBLK;
    stageBT(B, ldb, nb, kb, kcnt, lds);
    for (int k0 = 0; k0 < kcnt; k0 += 4) {
      const int ka = k0 + 2 * hi;
      const v2f a = *(const v2f*)(Arow + kb + ka);  // even offset in all callers
#pragma unroll
      for (int t = 0; t < 8; ++t) {
        const v2f b = *(const v2f*)(lds + (t * 16 + lm) * KP + ka);
        c[t] = wmma4(a, b, c[t]);
      }
    }
  }
}

// Fused row LayerNorm over the 16x128 rows held in C frags; x already includes
// residual+bias. shfl_xor offsets 8/4/2/1 stay within each 16-lane half = one
// row group, giving per-row mean/var broadcast to the owning lanes.
__device__ __forceinline__ void ln_store(v8f c[8], const float* __restrict__ g,
                                         const float* __restrict__ be,
                                         float* __restrict__ out, int m0,
                                         int hi, int lm) {
#pragma unroll
  for (int r = 0; r < 8; ++r) {
    float p = 0.f, p2 = 0.f;
#pragma unroll
    for (int t = 0; t < 8; ++t) {
      const float v = c[t][r];
      p += v;
      p2 += v * v;
    }
#pragma unroll
    for (int off = 8; off; off >>= 1) {
      p += __shfl_xor(p, off, 32);
      p2 += __shfl_xor(p2, off, 32);
    }
    const float mean = p * (1.0f / 128.0f);
    const float var = p2 * (1.0f / 128.0f) - mean * mean;
    const float rs = rsqrtf(var + 1e-5f);
    const int row = m0 + 8 * hi + r;
#pragma unroll
    for (int t = 0; t < 8; ++t) {
      const int col = t * 16 + lm;
      out[row * HID + col] = (c[t][r] - mean) * rs * g[col] + be[col];
    }
  }
}

// ---------------------------------------------------------------------------
// Rank-1 edge-embedding collapse: u_i = W_edge[0,:] @ We[i], v_i = b_edge @ We[i]
// ---------------------------------------------------------------------------
__global__ void k_uv(const float* __restrict__ We, const float* __restrict__ W_edge,
                     const float* __restrict__ b_edge, float* __restrict__ uv) {
  const int i = blockIdx.x;
  const int c = threadIdx.x;
  const float* W = We + i * HID * HID;
  float u = 0.f, v = 0.f;
  for (int k = 0; k < HID; ++k) {
    const float w = W[k * HID + c];
    u += W_edge[k] * w;
    v += b_edge[k] * w;
  }
  uv[i * 256 + c] = u;
  uv[i * 256 + 128 + c] = v;
}

// ---------------------------------------------------------------------------
// h = feats @ W_node + b_node + pos @ W_pos + b_pos   (K = 44 then K = 8)
// ---------------------------------------------------------------------------
__global__ void k_inproj(const float* __restrict__ feats,
                         const float* __restrict__ pos,
                         const float* __restrict__ Wn, const float* __restrict__ bn,
                         const float* __restrict__ Wp, const float* __restrict__ bp,
                         float* __restrict__ h) {
  __shared__ __align__(16) float lds[128 * KP];
  const int lane = threadIdx.x & 31, wv = threadIdx.x >> 5;
  const int hi = lane >> 4, lm = lane & 15;
  const int m0 = (blockIdx.x * 8 + wv) * 16;
  v8f c[8] = {};
  gemm_full(feats + (m0 + lm) * 44, Wn, HID, 0, 44, hi, lm, lds, c);
  gemm_full(pos + (m0 + lm) * 8, Wp, HID, 0, 8, hi, lm, lds, c);
#pragma unroll
  for (int t = 0; t < 8; ++t) {
    const int col = t * 16 + lm;
    const float bias = bn[col] + bp[col];
#pragma unroll
    for (int r = 0; r < 8; ++r)
      h[(m0 + 8 * hi + r) * HID + col] = c[t][r] + bias;
  }
}

// ---------------------------------------------------------------------------
// Q/K/V = h @ Wq/Wk/Wv  (blockIdx.y selects which; no bias in reference)
// ---------------------------------------------------------------------------
__global__ void k_qkv(const float* __restrict__ h,
                      const float* __restrict__ Wq, const float* __restrict__ Wk,
                      const float* __restrict__ Wv, float* __restrict__ Q,
                      float* __restrict__ K, float* __restrict__ V) {
  __shared__ __align__(16) float lds[128 * KP];
  const float* W = (blockIdx.y == 0) ? Wq : (blockIdx.y == 1) ? Wk : Wv;
  float* O = (blockIdx.y == 0) ? Q : (blockIdx.y == 1) ? K : V;
  const int lane = threadIdx.x & 31, wv = threadIdx.x >> 5;
  const int hi = lane >> 4, lm = lane & 15;
  const int m0 = (blockIdx.x * 8 + wv) * 16;
  v8f c[8] = {};
  gemm_full(h + (m0 + lm) * HID, W, HID, 0, HID, hi, lm, lds, c);
#pragma unroll
  for (int t = 0; t < 8; ++t) {
    const int col = t * 16 + lm;
#pragma unroll
    for (int r = 0; r < 8; ++r)
      O[(m0 + 8 * hi + r) * HID + col] = c[t][r];
  }
}

// ---------------------------------------------------------------------------
// Edge attention: one wave per edge, lane = channel within head (DK==32==wave32).
// ---------------------------------------------------------------------------
__global__ void k_edge(const float* __restrict__ Q, const float* __restrict__ K,
                       const float* __restrict__ V, const int* __restrict__ src,
                       const int* __restrict__ dst, const float* __restrict__ ef,
                       const float* __restrict__ u, const float* __restrict__ v,
                       float* __restrict__ Z, float* __restrict__ Hacc) {
  const int lane = threadIdx.x & 31;
  const int e = blockIdx.x * 8 + (threadIdx.x >> 5);
  const int s = src[e], d = dst[e];
  const float x = ef[e];
  const float* Ks = K + s * HID;
  const float* Qd = Q + d * HID;
  const float* Vs = V + s * HID;
#pragma unroll
  for (int hd = 0; hd < 4; ++hd) {
    const int c = hd * 32 + lane;
    float kq = Ks[c] * Qd[c];
#pragma unroll
    for (int off = 16; off; off >>= 1) kq += __shfl_xor(kq, off, 32);
    const float score = kq * 0.17677669529663689f;  // 1/sqrt(32)
    const float Ejc = x * u[c] + v[c];
    const float alpha = __expf(score * (1.0f + Ejc));
    atomicAdd(Z + d * HID + c, alpha);
    atomicAdd(Hacc + d * HID + c, Vs[c] * alpha);
  }
}

// ---------------------------------------------------------------------------
// h = LN1( h + (Hacc/max(Z,1e-6)) @ W_lin + b_lin )   (in-place on h)
// ---------------------------------------------------------------------------
__global__ void k_attn_ln(const float* __restrict__ Hacc, const float* __restrict__ Z,
                          const float* __restrict__ W, const float* __restrict__ b,
                          const float* __restrict__ g, const float* __restrict__ be,
                          float* __restrict__ h) {
  __shared__ __align__(16) float lds[128 * KP];
  const int lane = threadIdx.x & 31, wv = threadIdx.x >> 5;
  const int hi = lane >> 4, lm = lane & 15;
  const int m0 = (blockIdx.x * 8 + wv) * 16;
  const int row = m0 + lm;
  v8f c[8] = {};
  for (int kb = 0; kb < HID; kb += KBLK) {
    stageBT(W, HID, 0, kb, KBLK, lds);
    for (int k0 = 0; k0 < KBLK; k0 += 4) {
      const int ka = k0 + 2 * hi;
      const int kg = kb + ka;
      v2f a;
      a[0] = Hacc[row * HID + kg] / fmaxf(Z[row * HID + kg], 1e-6f);
      a[1] = Hacc[row * HID + kg + 1] / fmaxf(Z[row * HID + kg + 1], 1e-6f);
#pragma unroll
      for (int t = 0; t < 8; ++t) {
        const v2f bb = *(const v2f*)(lds + (t * 16 + lm) * KP + ka);
        c[t] = wmma4(a, bb, c[t]);
      }
    }
  }
#pragma unroll
  for (int t = 0; t < 8; ++t) {
    const int col = t * 16 + lm;
#pragma unroll
    for (int r = 0; r < 8; ++r) {
      const int rr = m0 + 8 * hi + r;
      c[t][r] = h[rr * HID + col] + c[t][r] + b[col];
    }
  }
  ln_store(c, g, be, h, m0, hi, lm);
}

// ---------------------------------------------------------------------------
// f = relu(h @ Wf1 + bf1)   (N=256 split across blockIdx.y)
// ---------------------------------------------------------------------------
__global__ void k_ffn1(const float* __restrict__ h, const float* __restrict__ W,
                       const float* __restrict__ b, float* __restrict__ f) {
  __shared__ __align__(16) float lds[128 * KP];
  const int lane = threadIdx.x & 31, wv = threadIdx.x >> 5;
  const int hi = lane >> 4, lm = lane & 15;
  const int m0 = (blockIdx.x * 8 + wv) * 16;
  const int nb = blockIdx.y * 128;
  v8f c[8] = {};
  gemm_full(h + (m0 + lm) * HID, W, 256, nb, HID, hi, lm, lds, c);
#pragma unroll
  for (int t = 0; t < 8; ++t) {
    const int col = nb + t * 16 + lm;
    const float bias = b[col];
#pragma unroll
    for (int r = 0; r < 8; ++r)
      f[(m0 + 8 * hi + r) * 256 + col] = fmaxf(c[t][r] + bias, 0.0f);
  }
}

// ---------------------------------------------------------------------------
// h = LN2( h + f @ Wf2 + bf2 )   (K = 256, in-place on h)
// ---------------------------------------------------------------------------
__global__ void k_ffn2_ln(const float* __restrict__ f, const float* __restrict__ W,
                          const float* __restrict__ b, const float* __restrict__ g,
                          const float* __restrict__ be, float* __restrict__ h) {
  __shared__ __align__(16) float lds[128 * KP];
  const int lane = threadIdx.x & 31, wv = threadIdx.x >> 5;
  const int hi = lane >> 4, lm = lane & 15;
  const int m0 = (blockIdx.x * 8 + wv) * 16;
  v8f c[8] = {};
  gemm_full(f + (m0 + lm) * 256, W, HID, 0, 256, hi, lm, lds, c);
#pragma unroll
  for (int t = 0; t < 8; ++t) {
    const int col = t * 16 + lm;
#pragma unroll
    for (int r = 0; r < 8; ++r) {
      const int rr = m0 + 8 * hi + r;
      c[t][r] = h[rr * HID + col] + c[t][r] + b[col];
    }
  }
  ln_store(c, g, be, h, m0, hi, lm);
}

// ---------------------------------------------------------------------------
// Deterministic two-stage mean over nodes.
// ---------------------------------------------------------------------------
__global__ void k_mean1(const float* __restrict__ h, float* __restrict__ part) {
  const int c = threadIdx.x;
  const float* p = h + (size_t)blockIdx.x * 256 * HID;
  float acc = 0.f;
  for (int r = 0; r < 256; ++r) acc += p[r * HID + c];
  part[blockIdx.x * HID + c] = acc;
}

__global__ void k_mean2(const float* __restrict__ part, float* __restrict__ out) {
  const int c = threadIdx.x;
  float acc = 0.f;
  for (int b = 0; b < 128; ++b) acc += part[b * HID + c];
  out[c] = acc * (1.0f / 32768.0f);
}

// ---------------------------------------------------------------------------
extern "C" void kernel_launch(void* const* d_in, const int* in_sizes, int n_in,
                              void* d_out, int out_size, void* d_ws, size_t ws_size,
                              hipStream_t stream) {
  const float* feats   = (const float*)d_in[0];
  const float* pos     = (const float*)d_in[1];
  const float* ef      = (const float*)d_in[2];
  const int*   src     = (const int*)d_in[3];
  const int*   dst     = (const int*)d_in[4];
  const float* W_node  = (const float*)d_in[5];
  const float* b_node  = (const float*)d_in[6];
  const float* W_pos   = (const float*)d_in[7];
  const float* b_pos   = (const float*)d_in[8];
  const float* W_edge  = (const float*)d_in[9];
  const float* b_edge  = (const float*)d_in[10];
  const float* Wq      = (const float*)d_in[11];
  const float* Wk      = (const float*)d_in[12];
  const float* Wv      = (const float*)d_in[13];
  const float* We      = (const float*)d_in[14];
  const float* W_lin   = (const float*)d_in[15];
  const float* b_lin   = (const float*)d_in[16];
  const float* g1      = (const float*)d_in[17];
  const float* be1     = (const float*)d_in[18];
  const float* Wf1     = (const float*)d_in[19];
  const float* bf1     = (const float*)d_in[20];
  const float* Wf2     = (const float*)d_in[21];
  const float* bf2     = (const float*)d_in[22];
  const float* g2      = (const float*)d_in[23];
  const float* be2     = (const float*)d_in[24];

  float* ws = (float*)d_ws;
  float* h    = ws;                    // 4M floats
  float* Qb   = ws + 4194304;          // 4M
  float* Kb   = ws + 8388608;          // 4M
  float* Vb   = ws + 12582912;         // 4M
  float* Zb   = ws + 16777216;         // 4M
  float* Ha   = ws + 20971520;         // 4M
  float* fb   = Qb;                    // FFN hidden (8M floats) reuses dead Q+K
  float* uv   = ws + 25165824;         // 3*256
  float* part = uv + 768;              // 128*128

  const dim3 blk(256);

  k_uv<<<dim3(3), dim3(128), 0, stream>>>(We, W_edge, b_edge, uv);
  k_inproj<<<dim3(256), blk, 0, stream>>>(feats, pos, W_node, b_node, W_pos, b_pos, h);

  for (int i = 0; i < 3; ++i) {
    k_qkv<<<dim3(256, 3), blk, 0, stream>>>(h, Wq + i * 16384, Wk + i * 16384,
                                            Wv + i * 16384, Qb, Kb, Vb);
    hipMemsetAsync(Zb, 0, (size_t)4194304 * sizeof(float), stream);
    hipMemsetAsync(Ha, 0, (size_t)4194304 * sizeof(float), stream);
    k_edge<<<dim3(65536), blk, 0, stream>>>(Qb, Kb, Vb, src, dst, ef,
                                            uv + i * 256, uv + i * 256 + 128, Zb, Ha);
    k_attn_ln<<<dim3(256), blk, 0, stream>>>(Ha, Zb, W_lin + i * 16384,
                                             b_lin + i * 128, g1 + i * 128,
                                             be1 + i * 128, h);
    k_ffn1<<<dim3(256, 2), blk, 0, stream>>>(h, Wf1 + i * 32768, bf1 + i * 256, fb);
    k_ffn2_ln<<<dim3(256), blk, 0, stream>>>(fb, Wf2 + i * 32768, bf2 + i * 128,
                                             g2 + i * 128, be2 + i * 128, h);
  }

  k_mean1<<<dim3(128), dim3(128), 0, stream>>>(h, part);
  k_mean2<<<dim3(1), dim3(128), 0, stream>>>(part, (float*)d_out);
}